// DeltaImportance_7524782702898
// MI455X (gfx1250) — compile-verified
//
#include <hip/hip_runtime.h>
#include <math.h>

constexpr int kB = 8;
constexpr int kT = 2048;
constexpr int kD = 512;

typedef __attribute__((ext_vector_type(2))) float v2f;
typedef __attribute__((ext_vector_type(8))) float v8f;

__device__ __forceinline__ float softplus_dev(const float* kp) {
    float k = kp[0];
    return fmaxf(k, 0.0f) + log1pf(expf(-fabsf(k)));   // softplus(k)
}

// -------- Pass 1: dsq[b,t] = sum_d (cumsum_t(h)[t,d] - h[t-1,d])^2 ---------
// One lane per d-column; running cumsum over T; wave-reduce + atomic add.
__global__ void delta_mag_sq_kernel(const float* __restrict__ h,
                                    float* __restrict__ dsq) {
    int gid  = blockIdx.x * blockDim.x + threadIdx.x;
    int lane = gid & 31;
    int wave = gid >> 5;                 // 0..127  (kB * kD/32)
    int b    = wave >> 4;                // 0..7
    int d    = ((wave & 15) << 5) + lane;// 0..511
    const float* hp = h + (size_t)b * kT * kD + d;

    float cum = 0.f, hprev = 0.f;
    for (int t = 0; t < kT; ++t) {
        float hv = hp[(size_t)t * kD];   // 128B coalesced across wave
        cum += hv;
        float dl = cum - hprev;          // delta = h_cum[t] - h[t-1]
        hprev = hv;
        float sq = dl * dl;
        #pragma unroll
        for (int off = 16; off > 0; off >>= 1)
            sq += __shfl_xor(sq, off, 32);
        if (lane == 0) atomicAdd(&dsq[b * kT + t], sq);
    }
}

// -------- Pass 1.5: dsq -> mag (in place), hoists sqrt off the scan path ---
__global__ void mag_sqrt_kernel(float* __restrict__ dsq) {
    int i = blockIdx.x * blockDim.x + threadIdx.x;  // kB*kT threads
    dsq[i] = sqrtf(dsq[i]);
}

// -------- Pass 2: chunked WMMA scan -----------------------------------------
// context[t] = h[t] + r*context[t-1]; per 16-chunk:
//   C_init[i][n] = r^(i+1) * S[n]   (carry via WMMA C operand)
//   C += L(16x16) * Hc(16x16)       (4x V_WMMA_F32_16X16X4_F32)
//   S_new[n] = C[15][n]             (row 15 == exact new carry)
// out[t][d] = mag[t] * |context[t][d]|
__global__ void context_wmma_kernel(const float* __restrict__ h,
                                    const float* __restrict__ kp,
                                    const float* __restrict__ mag,
                                    float* __restrict__ out) {
    const int lane  = threadIdx.x;        // blockDim.x == 32, EXEC all ones
    const int dtile = blockIdx.x & 31;    // 32 tiles of 16 along D
    const int b     = blockIdx.x >> 5;    // 0..7
    const int n     = lane & 15;          // column within d-tile (and M for A)
    const int hi    = (lane >> 4) & 1;    // lane half selects K/M offset

    const float sp  = softplus_dev(kp);
    const float l2r = -sp * 1.4426950408889634f;   // log2(r), r = exp(-sp)
    // r^dist without a dynamically-indexed table (avoids scratch spill):
    auto rpow = [&](int dist) -> float { return exp2f((float)dist * l2r); };

    // A tiles (loop-invariant): L[i][j] = (j<=i) ? r^(i-j) : 0
    // A layout 16x4 f32: lanes 0-15 -> M=i, VGPR p -> K = p + 2*hi
    v2f a[4];
    #pragma unroll
    for (int g = 0; g < 4; ++g) {
        int j0 = 4 * g + 2 * hi;
        a[g].x = (j0     <= n) ? rpow(n - j0)     : 0.f;
        a[g].y = (j0 + 1 <= n) ? rpow(n - j0 - 1) : 0.f;
    }

    // Per-lane carry-init powers: rpc[v] = r^(v + 1 + 8*hi)
    // (independent multiplies by S each chunk, no serial chain)
    float rpc[8];
    #pragma unroll
    for (int v = 0; v < 8; ++v) rpc[v] = rpow(v + 1 + 8 * hi);

    const float* hb = h   + ((size_t)b * kT) * kD + dtile * 16 + n;
    float*       ob = out + ((size_t)b * kT) * kD + dtile * 16 + n;
    const float* mb = mag + b * kT;

    float S = 0.f;                        // carry: context[t0-1][n]
    for (int t0 = 0; t0 < kT; t0 += 16) {
        // B tiles: B[k][n] = h[t0+k][d0+n]; lanes hold N, VGPR p -> K = p+2*hi
        v2f bm[4];
        #pragma unroll
        for (int g = 0; g < 4; ++g) {
            size_t j = (size_t)(t0 + 4 * g + 2 * hi);
            bm[g].x = hb[j * kD];
            bm[g].y = hb[(j + 1) * kD];
        }
        // C init with carry term: C[M][n] = r^(M+1)*S, M = v + 8*hi
        v8f c;
        #pragma unroll
        for (int v = 0; v < 8; ++v) c[v] = rpc[v] * S;

        c = __builtin_amdgcn_wmma_f32_16x16x4_f32(false, a[0], false, bm[0], (short)0, c, false, false);
        c = __builtin_amdgcn_wmma_f32_16x16x4_f32(false, a[1], false, bm[1], (short)0, c, false, false);
        c = __builtin_amdgcn_wmma_f32_16x16x4_f32(false, a[2], false, bm[2], (short)0, c, false, false);
        c = __builtin_amdgcn_wmma_f32_16x16x4_f32(false, a[3], false, bm[3], (short)0, c, false, false);

        // Store: out[t0+M][n] = mag[t0+M] * |context|
        #pragma unroll
        for (int v = 0; v < 8; ++v) {
            int M = v + 8 * hi;
            ob[(size_t)(t0 + M) * kD] = mb[t0 + M] * fabsf(c[v]);
        }
        // New carry = context row 15 (VGPR 7, lanes 16-31) -> broadcast to all
        S = __shfl(c[7], 16 + n, 32);
    }
}

extern "C" void kernel_launch(void* const* d_in, const int* in_sizes, int n_in,
                              void* d_out, int out_size, void* d_ws, size_t ws_size,
                              hipStream_t stream) {
    const float* h  = (const float*)d_in[0];
    const float* k  = (const float*)d_in[1];
    float* out = (float*)d_out;
    float* dsq = (float*)d_ws;            // kB*kT floats = 64 KB

    (void)hipMemsetAsync(dsq, 0, (size_t)kB * kT * sizeof(float), stream);
    delta_mag_sq_kernel<<<16, 256, 0, stream>>>(h, dsq);
    mag_sqrt_kernel<<<(kB * kT) / 256, 256, 0, stream>>>(dsq);
    context_wmma_kernel<<<kB * (kD / 16), 32, 0, stream>>>(h, k, dsq, out);
}